// TriplaneGrid_120259084882
// MI455X (gfx1250) — compile-verified
//
#include <hip/hip_runtime.h>
#include <cstddef>

#define RR 512
#define CC 32

typedef float v4f __attribute__((ext_vector_type(4)));

// ---------------------------------------------------------------------------
// Kernel 1: clip + transpose repack  [3, C, R, R] -> [3, R, R, C]
// Loads go global -> LDS via CDNA5 async-to-LDS (ASYNCcnt), per-lane LDS
// destination addresses land directly in the padded transpose tile.
// ---------------------------------------------------------------------------
__global__ __launch_bounds__(256) void repack_clip_kernel(const float* __restrict__ tri,
                                                          float* __restrict__ packed) {
    __shared__ float tile[32 * 33];  // [c][x], stride 33 -> conflict-free transpose read

    const int b  = blockIdx.x;       // ((p*512 + y)*16 + xt)
    const int xt = b & 15;
    const int y  = (b >> 4) & 511;
    const int p  = b >> 13;
    const int t  = threadIdx.x;

    // Async load phase: each wave (32 lanes) pulls one contiguous 128B row.
#pragma unroll
    for (int i = 0; i < 4; ++i) {
        const int c = (t >> 5) + i * 8;
        const int x = t & 31;
        const float* src = tri + ((size_t)((p * 32 + c) * 512 + y) * 512 + xt * 32 + x);
        unsigned lds_off = (unsigned)(size_t)(&tile[c * 33 + x]);
        asm volatile("global_load_async_to_lds_b32 %0, %1, off"
                     :: "v"(lds_off), "v"(src)
                     : "memory");
    }
    asm volatile("s_wait_asynccnt 0" ::: "memory");
    __syncthreads();

    // Transposed, clipped, coalesced store: consecutive lanes -> consecutive c.
    float* dstbase = packed + ((size_t)((p * 512 + y) * 512) + xt * 32) * 32;
#pragma unroll
    for (int i = 0; i < 4; ++i) {
        const int x = (t >> 5) + i * 8;
        const int c = t & 31;
        float v = tile[c * 33 + x];
        v = fminf(fmaxf(v, -1.0f), 1.0f);
        dstbase[(size_t)x * 32 + c] = v;
    }
}

// ---------------------------------------------------------------------------
// Kernel 2: thread-per-point gather from channels-last planes (L2-resident)
// ---------------------------------------------------------------------------
__device__ __forceinline__ void sample_plane(const v4f* __restrict__ plane,
                                             float px, float py, v4f acc[8]) {
    const float HALF = 0.5f * (RR - 1);
    const float u = (px + 1.0f) * HALF;
    const float v = (py + 1.0f) * HALF;
    const float fu = floorf(u), fv = floorf(v);
    const int ix0 = (int)fu, iy0 = (int)fv;
    const int ix1 = ix0 + 1, iy1 = iy0 + 1;
    const float ax = u - fu, ay = v - fv;

    const float bx0 = (ix0 >= 0 && ix0 < RR) ? 1.0f : 0.0f;
    const float bx1 = (ix1 >= 0 && ix1 < RR) ? 1.0f : 0.0f;
    const float by0 = (iy0 >= 0 && iy0 < RR) ? 1.0f : 0.0f;
    const float by1 = (iy1 >= 0 && iy1 < RR) ? 1.0f : 0.0f;
    const int cx0 = min(max(ix0, 0), RR - 1);
    const int cx1 = min(max(ix1, 0), RR - 1);
    const int cy0 = min(max(iy0, 0), RR - 1);
    const int cy1 = min(max(iy1, 0), RR - 1);

    const float w00 = (1.0f - ax) * (1.0f - ay) * bx0 * by0;
    const float w10 = ax * (1.0f - ay) * bx1 * by0;
    const float w01 = (1.0f - ax) * ay * bx0 * by1;
    const float w11 = ax * ay * bx1 * by1;

    const v4f* t00 = plane + ((size_t)cy0 * RR + cx0) * 8;
    const v4f* t10 = plane + ((size_t)cy0 * RR + cx1) * 8;
    const v4f* t01 = plane + ((size_t)cy1 * RR + cx0) * 8;
    const v4f* t11 = plane + ((size_t)cy1 * RR + cx1) * 8;

#pragma unroll
    for (int k = 0; k < 8; ++k) acc[k] += w00 * t00[k];
#pragma unroll
    for (int k = 0; k < 8; ++k) acc[k] += w10 * t10[k];
#pragma unroll
    for (int k = 0; k < 8; ++k) acc[k] += w01 * t01[k];
#pragma unroll
    for (int k = 0; k < 8; ++k) acc[k] += w11 * t11[k];
}

__global__ __launch_bounds__(256) void triplane_gather_kernel(const float* __restrict__ coords,
                                                              const float* __restrict__ packed,
                                                              float* __restrict__ out, int M) {
    const int m = blockIdx.x * 256 + threadIdx.x;
    if (m >= M) return;

    // Coords are streamed once: NT loads, keep L2 for the planes.
    const float cx = __builtin_nontemporal_load(coords + 3 * (size_t)m + 0);
    const float cy = __builtin_nontemporal_load(coords + 3 * (size_t)m + 1);
    const float cz = __builtin_nontemporal_load(coords + 3 * (size_t)m + 2);

    v4f acc[8];
#pragma unroll
    for (int k = 0; k < 8; ++k) acc[k] = (v4f)0.0f;

    const v4f* base = (const v4f*)packed;
    sample_plane(base + (size_t)0 * RR * RR * 8, cy, cx, acc);  // plane 0: (y, x)
    sample_plane(base + (size_t)1 * RR * RR * 8, cz, cx, acc);  // plane 1: (z, x)
    sample_plane(base + (size_t)2 * RR * RR * 8, cy, cz, acc);  // plane 2: (y, z)

    // Output written once: non-temporal b128 stores, don't pollute L2.
    v4f* o = (v4f*)out + (size_t)m * 8;
#pragma unroll
    for (int k = 0; k < 8; ++k) __builtin_nontemporal_store(acc[k], o + k);
}

// ---------------------------------------------------------------------------
// Fallback: gather directly from [3, C, R, R] if workspace too small.
// ---------------------------------------------------------------------------
__global__ __launch_bounds__(256) void triplane_gather_direct(const float* __restrict__ coords,
                                                              const float* __restrict__ tri,
                                                              float* __restrict__ out, int M) {
    const int m = blockIdx.x * 256 + threadIdx.x;
    if (m >= M) return;
    const float cx = coords[3 * (size_t)m + 0];
    const float cy = coords[3 * (size_t)m + 1];
    const float cz = coords[3 * (size_t)m + 2];

    float acc[CC];
#pragma unroll
    for (int c = 0; c < CC; ++c) acc[c] = 0.0f;

    const float pxs[3] = {cy, cz, cy};
    const float pys[3] = {cx, cx, cz};
    const float HALF = 0.5f * (RR - 1);

    for (int p = 0; p < 3; ++p) {
        const float u = (pxs[p] + 1.0f) * HALF;
        const float v = (pys[p] + 1.0f) * HALF;
        const float fu = floorf(u), fv = floorf(v);
        const int ix0 = (int)fu, iy0 = (int)fv;
        const float ax = u - fu, ay = v - fv;
        const int xs[2] = {min(max(ix0, 0), RR - 1), min(max(ix0 + 1, 0), RR - 1)};
        const int ys[2] = {min(max(iy0, 0), RR - 1), min(max(iy0 + 1, 0), RR - 1)};
        const float vx[2] = {(ix0 >= 0 && ix0 < RR) ? 1.0f : 0.0f,
                             (ix0 + 1 >= 0 && ix0 + 1 < RR) ? 1.0f : 0.0f};
        const float vy[2] = {(iy0 >= 0 && iy0 < RR) ? 1.0f : 0.0f,
                             (iy0 + 1 >= 0 && iy0 + 1 < RR) ? 1.0f : 0.0f};
        const float wx[2] = {(1.0f - ax) * vx[0], ax * vx[1]};
        const float wy[2] = {(1.0f - ay) * vy[0], ay * vy[1]};
        for (int j = 0; j < 2; ++j) {
            for (int i = 0; i < 2; ++i) {
                const float w = wx[i] * wy[j];
                const float* tp = tri + ((size_t)(p * 32) * RR * RR) + (size_t)ys[j] * RR + xs[i];
#pragma unroll
                for (int c = 0; c < CC; ++c) {
                    float val = tp[(size_t)c * RR * RR];
                    val = fminf(fmaxf(val, -1.0f), 1.0f);
                    acc[c] += w * val;
                }
            }
        }
    }
#pragma unroll
    for (int c = 0; c < CC; ++c)
        __builtin_nontemporal_store(acc[c], out + (size_t)m * CC + c);
}

// ---------------------------------------------------------------------------
extern "C" void kernel_launch(void* const* d_in, const int* in_sizes, int n_in,
                              void* d_out, int out_size, void* d_ws, size_t ws_size,
                              hipStream_t stream) {
    const float* coords = (const float*)d_in[0];
    const float* tri    = (const float*)d_in[1];
    float* out          = (float*)d_out;
    const int M = in_sizes[0] / 3;

    const size_t need = (size_t)3 * RR * RR * CC * sizeof(float);  // 100.7 MB
    if (ws_size >= need) {
        float* packed = (float*)d_ws;
        repack_clip_kernel<<<3 * 512 * 16, 256, 0, stream>>>(tri, packed);
        triplane_gather_kernel<<<(M + 255) / 256, 256, 0, stream>>>(coords, packed, out, M);
    } else {
        triplane_gather_direct<<<(M + 255) / 256, 256, 0, stream>>>(coords, tri, out, M);
    }
}